// GAT_9285719294177
// MI455X (gfx1250) — compile-verified
//
#include <hip/hip_runtime.h>
#include <hip/hip_bf16.h>

// ---------------- problem constants (from reference) ----------------
#define NND   50000      // nodes
#define FIN   128        // input features
#define NH    8          // heads, layer 1
#define HIDC  32         // channels per head, layer 1
#define HC    256        // NH*HIDC
#define ODIM  40         // layer-2 output dim
#define NE    800000     // edges (without self loops)
#define NET   850000     // edges + self loops
#define NEG_SLOPE 0.2f
#define EPSV  1e-16f

typedef __attribute__((ext_vector_type(2))) float v2f;
typedef __attribute__((ext_vector_type(8))) float v8f;

// ---------------- helpers ----------------
__device__ __forceinline__ float wave_sum(float v) {
#pragma unroll
  for (int off = 16; off > 0; off >>= 1) v += __shfl_xor(v, off, 32);
  return v;
}
__device__ __forceinline__ int f32_ord(float x) {
  int i = __float_as_int(x);
  return i >= 0 ? i : (i ^ 0x7FFFFFFF);
}
__device__ __forceinline__ float ord_f32(int i) {
  return __int_as_float(i >= 0 ? i : (i ^ 0x7FFFFFFF));
}

// ---------------- fills ----------------
__global__ void k_fill_f32(float* __restrict__ p, float v, int n) {
  int i = blockIdx.x * blockDim.x + threadIdx.x;
  if (i < n) p[i] = v;
}
__global__ void k_fill_i32(int* __restrict__ p, int v, int n) {
  int i = blockIdx.x * blockDim.x + threadIdx.x;
  if (i < n) p[i] = v;
}

// ---------------- GEMM1: h1[50000,256] = x[50000,128] @ W1[128,256] ----------------
// grid (3125, 2), block 256 (8 waves). Wave w does N-tile nhalf*128 + w*16.
// A tile (16x128) staged in LDS; fp32 WMMA 16x16x4.
__global__ __launch_bounds__(256) void k_gemm1(const float* __restrict__ x,
                                               const float* __restrict__ W,
                                               float* __restrict__ h) {
  __shared__ float As[16 * FIN];                // 8 KB
  const int tid  = threadIdx.x;
  const int wave = tid >> 5, lane = tid & 31;
  const int m0   = blockIdx.x * 16;
  {   // 2048 floats = 512 float4, 256 threads x 2
    const float4* xv = (const float4*)(x + (size_t)m0 * FIN);
    float4* sv = (float4*)As;
    sv[tid]       = xv[tid];
    sv[tid + 256] = xv[tid + 256];
  }
  __syncthreads();
  const int n0 = blockIdx.y * 128 + wave * 16;
  const int la = lane & 15, hi = lane >> 4;
  v8f acc = {};
#pragma unroll
  for (int k0 = 0; k0 < FIN; k0 += 4) {
    const int ka = k0 + hi * 2;
    v2f a, b;
    a.x = As[la * FIN + ka];
    a.y = As[la * FIN + ka + 1];
    b.x = W[ka * HC + n0 + la];
    b.y = W[(ka + 1) * HC + n0 + la];
    acc = __builtin_amdgcn_wmma_f32_16x16x4_f32(false, a, false, b, (short)0,
                                                acc, false, false);
  }
#pragma unroll
  for (int j = 0; j < 8; ++j)
    h[(size_t)(m0 + j + hi * 8) * HC + n0 + la] = acc[j];
}

// ---------------- GEMM2: t2[50000,40] = h2[50000,256] @ W2[256,40] ----------------
// grid 3125, block 96 (3 waves); N padded 40->48, guarded.
__global__ __launch_bounds__(96) void k_gemm2(const float* __restrict__ h2,
                                              const float* __restrict__ W,
                                              float* __restrict__ t) {
  __shared__ float As[16 * HC];                 // 16 KB
  const int tid  = threadIdx.x;
  const int wave = tid >> 5, lane = tid & 31;
  const int m0   = blockIdx.x * 16;
  {   // 4096 floats = 1024 float4, 96 threads strided
    const float4* xv = (const float4*)(h2 + (size_t)m0 * HC);
    float4* sv = (float4*)As;
    for (int i = tid; i < 1024; i += 96) sv[i] = xv[i];
  }
  __syncthreads();
  const int n0 = wave * 16;
  const int la = lane & 15, hi = lane >> 4;
  const int col = n0 + la;
  const bool ok = col < ODIM;
  v8f acc = {};
#pragma unroll 8
  for (int k0 = 0; k0 < HC; k0 += 4) {
    const int ka = k0 + hi * 2;
    v2f a, b;
    a.x = As[la * HC + ka];
    a.y = As[la * HC + ka + 1];
    b.x = ok ? W[ka * ODIM + col] : 0.f;
    b.y = ok ? W[(ka + 1) * ODIM + col] : 0.f;
    acc = __builtin_amdgcn_wmma_f32_16x16x4_f32(false, a, false, b, (short)0,
                                                acc, false, false);
  }
#pragma unroll
  for (int j = 0; j < 8; ++j)
    if (ok) t[(size_t)(m0 + j + hi * 8) * ODIM + col] = acc[j];
}

// ---------------- attention dot products ----------------
// layer1: wave per node; per head, 32-lane dot with att vectors.
__global__ __launch_bounds__(256) void k_att1(const float* __restrict__ h,
                                              const float* __restrict__ as_,
                                              const float* __restrict__ ad_,
                                              float* __restrict__ os,
                                              float* __restrict__ od) {
  const int node = blockIdx.x * 8 + (threadIdx.x >> 5);
  const int lane = threadIdx.x & 31;
  if (node >= NND) return;
  const float* hr = h + (size_t)node * HC;
#pragma unroll
  for (int hd = 0; hd < NH; ++hd) {
    const float v = hr[hd * HIDC + lane];
    const float s = wave_sum(v * as_[hd * HIDC + lane]);
    const float d = wave_sum(v * ad_[hd * HIDC + lane]);
    if (lane == 0) { os[node * NH + hd] = s; od[node * NH + hd] = d; }
  }
}

// layer2: wave per node over 40 channels (H=1).
__global__ __launch_bounds__(256) void k_att2(const float* __restrict__ t,
                                              const float* __restrict__ as_,
                                              const float* __restrict__ ad_,
                                              float* __restrict__ os,
                                              float* __restrict__ od) {
  const int node = blockIdx.x * 8 + (threadIdx.x >> 5);
  const int lane = threadIdx.x & 31;
  if (node >= NND) return;
  const float* tr = t + (size_t)node * ODIM;
  float s = 0.f, d = 0.f;
  {
    const float v = tr[lane];
    s += v * as_[lane]; d += v * ad_[lane];
  }
  if (lane < ODIM - 32) {
    const float v = tr[32 + lane];
    s += v * as_[32 + lane]; d += v * ad_[32 + lane];
  }
  s = wave_sum(s); d = wave_sum(d);
  if (lane == 0) { os[node] = s; od[node] = d; }
}

// ---------------- edge passes, layer 1 (H=8) ----------------
__device__ __forceinline__ void edge_sd(const int* ei, int eid, int& s, int& d) {
  if (eid < NE) { s = ei[eid]; d = ei[NE + eid]; }
  else          { s = d = eid - NE; }
}

__global__ void k_edgeA1(const int* __restrict__ ei,
                         const float* __restrict__ as_, const float* __restrict__ ad_,
                         float* __restrict__ e, int* __restrict__ nmax) {
  const int t = blockIdx.x * blockDim.x + threadIdx.x;
  if (t >= NET * NH) return;
  const int eid = t >> 3, hd = t & 7;
  int s, d; edge_sd(ei, eid, s, d);
  float v = as_[s * NH + hd] + ad_[d * NH + hd];
  v = v > 0.f ? v : v * NEG_SLOPE;
  e[t] = v;
  atomicMax(&nmax[d * NH + hd], f32_ord(v));
}

__global__ void k_edgeB1(const int* __restrict__ ei, float* __restrict__ e,
                         const int* __restrict__ nmax, float* __restrict__ denom) {
  const int t = blockIdx.x * blockDim.x + threadIdx.x;
  if (t >= NET * NH) return;
  const int eid = t >> 3, hd = t & 7;
  int s, d; edge_sd(ei, eid, s, d);
  const float ex = __expf(e[t] - ord_f32(nmax[d * NH + hd]));
  e[t] = ex;
  atomicAdd(&denom[d * NH + hd], ex);
}

// wave per edge: gather 1KB contiguous h[src] row, scatter-add 256 channels.
__global__ __launch_bounds__(256) void k_edgeC1(const int* __restrict__ ei,
                                                const float* __restrict__ h,
                                                const float* __restrict__ eexp,
                                                const float* __restrict__ denom,
                                                float* __restrict__ agg) {
  const int eid  = blockIdx.x * 8 + (threadIdx.x >> 5);
  const int lane = threadIdx.x & 31;
  if (eid >= NET) return;
  int s, d; edge_sd(ei, eid, s, d);
  const float* hs = h + (size_t)s * HC;
  float* ag = agg + (size_t)d * HC;
  __builtin_prefetch(hs, 0, 0);          // global_prefetch_b8
#pragma unroll
  for (int hd = 0; hd < NH; ++hd) {
    const float alpha = eexp[(size_t)eid * NH + hd] /
                        (denom[d * NH + hd] + EPSV);
    atomicAdd(&ag[hd * HIDC + lane], hs[hd * HIDC + lane] * alpha);
  }
}

// ---------------- edge passes, layer 2 (H=1) ----------------
__global__ void k_edgeA2(const int* __restrict__ ei,
                         const float* __restrict__ as_, const float* __restrict__ ad_,
                         float* __restrict__ e, int* __restrict__ nmax) {
  const int eid = blockIdx.x * blockDim.x + threadIdx.x;
  if (eid >= NET) return;
  int s, d; edge_sd(ei, eid, s, d);
  float v = as_[s] + ad_[d];
  v = v > 0.f ? v : v * NEG_SLOPE;
  e[eid] = v;
  atomicMax(&nmax[d], f32_ord(v));
}

__global__ void k_edgeB2(const int* __restrict__ ei, float* __restrict__ e,
                         const int* __restrict__ nmax, float* __restrict__ denom) {
  const int eid = blockIdx.x * blockDim.x + threadIdx.x;
  if (eid >= NET) return;
  int s, d; edge_sd(ei, eid, s, d);
  const float ex = __expf(e[eid] - ord_f32(nmax[d]));
  e[eid] = ex;
  atomicAdd(&denom[d], ex);
}

__global__ __launch_bounds__(256) void k_edgeC2(const int* __restrict__ ei,
                                                const float* __restrict__ t,
                                                const float* __restrict__ eexp,
                                                const float* __restrict__ denom,
                                                float* __restrict__ agg) {
  const int eid  = blockIdx.x * 8 + (threadIdx.x >> 5);
  const int lane = threadIdx.x & 31;
  if (eid >= NET) return;
  int s, d; edge_sd(ei, eid, s, d);
  const float alpha = eexp[eid] / (denom[d] + EPSV);
  const float* ts = t + (size_t)s * ODIM;
  float* ag = agg + (size_t)d * ODIM;
  atomicAdd(&ag[lane], ts[lane] * alpha);
  if (lane < ODIM - 32) atomicAdd(&ag[32 + lane], ts[32 + lane] * alpha);
}

// ---------------- elementwise epilogues ----------------
__global__ void k_elu_bias(float* __restrict__ a, const float* __restrict__ b) {
  const int i = blockIdx.x * blockDim.x + threadIdx.x;
  if (i >= NND * HC) return;
  const float v = a[i] + b[i & (HC - 1)];
  a[i] = v > 0.f ? v : (__expf(v) - 1.f);
}

__global__ void k_out_bias(const float* __restrict__ agg, const float* __restrict__ b,
                           float* __restrict__ out) {
  const int i = blockIdx.x * blockDim.x + threadIdx.x;
  if (i >= NND * ODIM) return;
  out[i] = agg[i] + b[i % ODIM];
}

// ---------------- host launcher ----------------
extern "C" void kernel_launch(void* const* d_in, const int* in_sizes, int n_in,
                              void* d_out, int out_size, void* d_ws, size_t ws_size,
                              hipStream_t stream) {
  const float* x   = (const float*)d_in[0];
  const int*   ei  = (const int*)d_in[1];
  const float* W1  = (const float*)d_in[2];
  const float* as1 = (const float*)d_in[3];
  const float* ad1 = (const float*)d_in[4];
  const float* b1  = (const float*)d_in[5];
  const float* W2  = (const float*)d_in[6];
  const float* as2 = (const float*)d_in[7];
  const float* ad2 = (const float*)d_in[8];
  const float* b2  = (const float*)d_in[9];
  float* out = (float*)d_out;

  // workspace layout (floats)
  float* ws   = (float*)d_ws;
  float* h1   = ws;                        // 12,800,000
  float* agg1 = h1   + (size_t)NND * HC;   // 12,800,000
  float* aS1  = agg1 + (size_t)NND * HC;   //    400,000
  float* aD1  = aS1  + NND * NH;
  int*   nm1  = (int*)(aD1 + NND * NH);
  float* dn1  = (float*)(nm1 + NND * NH);
  float* e1   = dn1  + NND * NH;           //  6,800,000
  float* t2   = e1   + (size_t)NET * NH;   //  2,000,000
  float* aS2  = t2   + (size_t)NND * ODIM;
  float* aD2  = aS2  + NND;
  int*   nm2  = (int*)(aD2 + NND);
  float* dn2  = (float*)(nm2 + NND);
  float* e2   = dn2  + NND;                //    850,000
  float* agg2 = e2   + NET;                //  2,000,000

  const int B = 256;
  // ---- init accumulators ----
  k_fill_f32<<<(NND * HC + B - 1) / B, B, 0, stream>>>(agg1, 0.f, NND * HC);
  k_fill_f32<<<(NND * NH + B - 1) / B, B, 0, stream>>>(dn1, 0.f, NND * NH);
  k_fill_i32<<<(NND * NH + B - 1) / B, B, 0, stream>>>(nm1, (int)0x80000000, NND * NH);
  k_fill_f32<<<(NND * ODIM + B - 1) / B, B, 0, stream>>>(agg2, 0.f, NND * ODIM);
  k_fill_f32<<<(NND + B - 1) / B, B, 0, stream>>>(dn2, 0.f, NND);
  k_fill_i32<<<(NND + B - 1) / B, B, 0, stream>>>(nm2, (int)0x80000000, NND);

  // ---- layer 1 ----
  k_gemm1<<<dim3(NND / 16, 2), 256, 0, stream>>>(x, W1, h1);
  k_att1<<<(NND + 7) / 8, 256, 0, stream>>>(h1, as1, ad1, aS1, aD1);
  k_edgeA1<<<(NET * NH + B - 1) / B, B, 0, stream>>>(ei, aS1, aD1, e1, nm1);
  k_edgeB1<<<(NET * NH + B - 1) / B, B, 0, stream>>>(ei, e1, nm1, dn1);
  k_edgeC1<<<(NET + 7) / 8, 256, 0, stream>>>(ei, h1, e1, dn1, agg1);
  k_elu_bias<<<(NND * HC + B - 1) / B, B, 0, stream>>>(agg1, b1);

  // ---- layer 2 ----
  k_gemm2<<<NND / 16, 96, 0, stream>>>(agg1, W2, t2);
  k_att2<<<(NND + 7) / 8, 256, 0, stream>>>(t2, as2, ad2, aS2, aD2);
  k_edgeA2<<<(NET + B - 1) / B, B, 0, stream>>>(ei, aS2, aD2, e2, nm2);
  k_edgeB2<<<(NET + B - 1) / B, B, 0, stream>>>(ei, e2, nm2, dn2);
  k_edgeC2<<<(NET + 7) / 8, 256, 0, stream>>>(ei, t2, e2, dn2, agg2);
  k_out_bias<<<(NND * ODIM + B - 1) / B, B, 0, stream>>>(agg2, b2, out);
}